// WaveletDomainFeature_79860621902106
// MI455X (gfx1250) — compile-verified
//
#include <hip/hip_runtime.h>
#include <stdint.h>

// ---------------------------------------------------------------------------
// WaveletDomainFeature for MI455X (gfx1250, wave32)
//   pass 1: db4 DWT detail-band abs-sum  [96][512][512] -> freq [96][259][259]
//           - async global->LDS gather (symmetric extension baked into addrs)
//           - horizontal conv in VALU, vertical conv as banded f32 WMMA
//             (A-taps generated branchlessly from inline literals)
//   pass 2: half-pixel bilinear resize 259->512, non-temporal output stores
//           (freq stays L2-resident: 26 MB << 192 MB L2)
// ---------------------------------------------------------------------------

#define IN_H   512
#define IN_W   512
#define OUT_HW 259
#define NPLANE 96
#define PLANE_IN  (IN_H * IN_W)
#define PLANE_FR  (OUT_HW * OUT_HW)

typedef float v2f __attribute__((ext_vector_type(2)));
typedef float v4f __attribute__((ext_vector_type(4)));
typedef float v8f __attribute__((ext_vector_type(8)));

// time-reversed db4 decomposition filters (correlation kernels), as literals
#define LO0  0.23037781330885523f
#define LO1  0.7148465705525415f
#define LO2  0.6308807679295904f
#define LO3 -0.02798376941698385f
#define LO4 -0.18703481171888114f
#define LO5  0.030841381835986965f
#define LO6  0.032883011666982945f
#define LO7 -0.010597401784997278f
#define HI0 -0.010597401784997278f
#define HI1 -0.032883011666982945f
#define HI2  0.030841381835986965f
#define HI3  0.18703481171888114f
#define HI4 -0.02798376941698385f
#define HI5 -0.6308807679295904f
#define HI6  0.7148465705525415f
#define HI7 -0.23037781330885523f

__device__ __constant__ float LO_R[8] = {LO0,LO1,LO2,LO3,LO4,LO5,LO6,LO7};
__device__ __constant__ float HI_R[8] = {HI0,HI1,HI2,HI3,HI4,HI5,HI6,HI7};

// symmetric (half-sample) extension index map, valid for e in [-512, 1023]
__device__ __forceinline__ int msym(int e) {
    e = (e < 0) ? (-1 - e) : e;
    e = (e >= IN_H) ? (2 * IN_H - 1 - e) : e;
    return e;
}

__global__ __launch_bounds__(128)
void dwt_detail_kernel(const float* __restrict__ x, float* __restrict__ freq) {
    __shared__ float s_in[38][136];   // 38 x 134 input patch (padded stride)
    __shared__ float s_lo[40][68];    // low-pass along W (rows 38,39 zeroed)
    __shared__ float s_hi[40][68];    // high-pass along W (rows 38,39 zeroed)

    const int wt = blockIdx.x;        // 0..4  -> w' tile of 64
    const int ht = blockIdx.y;        // 0..16 -> h' tile of 16
    const int p  = blockIdx.z;        // 0..95 plane
    const int h0 = ht * 16;
    const int w0 = wt * 64;
    const int tid = threadIdx.x;

    const float* src = x + (size_t)p * PLANE_IN;

    // ---- async global -> LDS gather of the 38x134 patch -------------------
    // extended coords: eh = 2*h0-6+i (i<38), ew = 2*w0-6+c (c<134)
    for (int it = 0; it < 40; ++it) {
        int q = tid + it * 128;
        if (q < 38 * 134) {
            int i = q / 134;
            int c = q - i * 134;
            const float* gp = src + msym(2 * h0 - 6 + i) * IN_W + msym(2 * w0 - 6 + c);
            unsigned lds = (unsigned)(uintptr_t)&s_in[i][c];     // low 32 bits = LDS offset
            unsigned long long ga = (unsigned long long)(uintptr_t)gp;
            asm volatile("global_load_async_to_lds_b32 %0, %1, off"
                         :: "v"(lds), "v"(ga) : "memory");
        }
    }
    asm volatile("s_wait_asynccnt 0x0" ::: "memory");
    __syncthreads();

    // ---- stage 1: horizontal stride-2 correlation (lo & hi) ---------------
    // out col c' uses s_in[i][2c'+j], j=0..7 ; 38*64 = 2432 = 19 * 128
    for (int it = 0; it < 19; ++it) {
        int q = tid + it * 128;
        int i = q >> 6;
        int c = q & 63;
        float lo = 0.0f, hi = 0.0f;
#pragma unroll
        for (int j = 0; j < 8; ++j) {
            float v = s_in[i][2 * c + j];
            lo += LO_R[j] * v;   // literal j -> folds to immediates
            hi += HI_R[j] * v;
        }
        s_lo[i][c] = lo;
        s_hi[i][c] = hi;
    }
    // zero-fill pad rows 38,39 so stage-2 B reads need no guards
    s_lo[38 + (tid >> 6)][tid & 63] = 0.0f;
    s_hi[38 + (tid >> 6)][tid & 63] = 0.0f;
    __syncthreads();

    // ---- stage 2: vertical conv as banded matmul on V_WMMA_F32_16X16X4_F32
    // out[m][n] = sum_k W[m][k] * z[k][n],  W[m][k] = f[k - 2m] (8-wide band)
    // K = 40 in ten K=4 slices. Per lane the A index 4s+kl-2m hits the 8-wide
    // window for exactly two slices sA, sA+1 with taps f[p], f[p+4].
    const int lane = tid & 31;
    const int wv   = tid >> 5;        // 4 waves -> 16-col strips
    const int m    = lane & 15;       // M index
    const int hih  = lane >> 4;       // 0: K slots {0,1}, 1: K slots {2,3}
    const int col  = wv * 16 + m;     // local column for B reads

    const int b0  = 2 * hih - 2 * m;  // even ; A.x index base
    const int b1  = b0 + 1;           // odd  ; A.y index base
    const int px  = b0 & 3;           // 0 or 2
    const int py  = b1 & 3;           // 1 or 3
    const int sAx = (px - b0) >> 2;   // in [0,8]
    const int sAy = (py - b1) >> 2;   // in [0,8]
    // tap pairs, one v_cndmask each
    const float xlo0 = (px == 0) ? LO0 : LO2;
    const float xlo1 = (px == 0) ? LO4 : LO6;
    const float xhi0 = (px == 0) ? HI0 : HI2;
    const float xhi1 = (px == 0) ? HI4 : HI6;
    const float ylo0 = (py == 1) ? LO1 : LO3;
    const float ylo1 = (py == 1) ? LO5 : LO7;
    const float yhi0 = (py == 1) ? HI1 : HI3;
    const float yhi1 = (py == 1) ? HI5 : HI7;

    v8f alh = {};
    v8f ahl = {};
    v8f ahh = {};

#pragma unroll
    for (int s = 0; s < 10; ++s) {
        v2f aLo, aHi, bLo, bHi;
        aLo.x = (s == sAx) ? xlo0 : ((s == sAx + 1) ? xlo1 : 0.0f);
        aHi.x = (s == sAx) ? xhi0 : ((s == sAx + 1) ? xhi1 : 0.0f);
        aLo.y = (s == sAy) ? ylo0 : ((s == sAy + 1) ? ylo1 : 0.0f);
        aHi.y = (s == sAy) ? yhi0 : ((s == sAy + 1) ? yhi1 : 0.0f);

        const int kx = 4 * s + 2 * hih;   // < 40, pad rows are zero
        bLo.x = s_lo[kx][col];
        bLo.y = s_lo[kx + 1][col];
        bHi.x = s_hi[kx][col];
        bHi.y = s_hi[kx + 1][col];

        // LH = hi(H) o lo(W), HL = lo(H) o hi(W), HH = hi(H) o hi(W)
        alh = __builtin_amdgcn_wmma_f32_16x16x4_f32(false, aHi, false, bLo,
                                                    (short)0, alh, false, false);
        ahl = __builtin_amdgcn_wmma_f32_16x16x4_f32(false, aLo, false, bHi,
                                                    (short)0, ahl, false, false);
        ahh = __builtin_amdgcn_wmma_f32_16x16x4_f32(false, aHi, false, bHi,
                                                    (short)0, ahh, false, false);
    }

    // D layout: VGPR v holds M=v (lanes 0-15) / M=8+v (lanes 16-31), N=lane%16
    float* fp = freq + (size_t)p * PLANE_FR;
    const int gcol = w0 + col;
#pragma unroll
    for (int v = 0; v < 8; ++v) {
        int grow = h0 + v + 8 * hih;
        if (grow < OUT_HW && gcol < OUT_HW) {
            float r = fabsf(alh[v]) + fabsf(ahl[v]) + fabsf(ahh[v]);
            fp[grow * OUT_HW + gcol] = r;
        }
    }
}

// ---------------------------------------------------------------------------
// pass 2: bilinear (half-pixel, align_corners=False) 259 -> 512
// ---------------------------------------------------------------------------
__global__ __launch_bounds__(256)
void bilinear_kernel(const float* __restrict__ freq, float* __restrict__ out) {
    const float scale = (float)OUT_HW / (float)IN_H;   // 259/512
    int idx = blockIdx.x * 256 + threadIdx.x;          // 4 outputs per thread
    int p   = idx / (IN_H * (IN_W / 4));
    int rem = idx - p * (IN_H * (IN_W / 4));
    int oy  = rem >> 7;
    int ox0 = (rem & 127) << 2;

    float sy = (oy + 0.5f) * scale - 0.5f;
    sy = fmaxf(sy, 0.0f);
    int y0 = (int)sy;
    y0 = min(y0, OUT_HW - 1);
    int y1 = min(y0 + 1, OUT_HW - 1);
    float fy = sy - (float)y0;

    const float* r0 = freq + (size_t)p * PLANE_FR + y0 * OUT_HW;
    const float* r1 = freq + (size_t)p * PLANE_FR + y1 * OUT_HW;

    v4f res;
#pragma unroll
    for (int d = 0; d < 4; ++d) {
        int ox = ox0 + d;
        float sx = (ox + 0.5f) * scale - 0.5f;
        sx = fmaxf(sx, 0.0f);
        int x0 = (int)sx;
        x0 = min(x0, OUT_HW - 1);
        int x1 = min(x0 + 1, OUT_HW - 1);
        float fx = sx - (float)x0;
        float top = r0[x0] + fx * (r0[x1] - r0[x0]);
        float bot = r1[x0] + fx * (r1[x1] - r1[x0]);
        res[d] = top + fy * (bot - top);
    }
    // streaming store: keep freq resident in L2 while writing 96 MiB out
    __builtin_nontemporal_store(res, (v4f*)(out + (size_t)p * PLANE_IN + oy * IN_W + ox0));
}

extern "C" void kernel_launch(void* const* d_in, const int* in_sizes, int n_in,
                              void* d_out, int out_size, void* d_ws, size_t ws_size,
                              hipStream_t stream) {
    const float* x   = (const float*)d_in[0];
    float*       out = (float*)d_out;
    float*       fr  = (float*)d_ws;          // needs 96*259*259*4 B ~ 24.6 MiB

    dim3 g1(5, 17, NPLANE);                   // 5 w-tiles(64) x 17 h-tiles(16) x 96 planes
    hipLaunchKernelGGL(dwt_detail_kernel, g1, dim3(128), 0, stream, x, fr);

    int blocks2 = (NPLANE * IN_H * (IN_W / 4)) / 256;   // 24576
    hipLaunchKernelGGL(bilinear_kernel, dim3(blocks2), dim3(256), 0, stream, fr, out);
}